// SparseVectorizedLiquidGraph_11596411699830
// MI455X (gfx1250) — compile-verified
//
#include <hip/hip_runtime.h>
#include <stdint.h>
#include <math.h>

#define INPUT_SIZE   1024
#define HIDDEN_SIZE  8192
#define OUTPUT_SIZE  256
#define PARAM_SIZE   2
#define BATCH        16
#define DT_STEP      0.1f
#define LO_IDX       (HIDDEN_SIZE - OUTPUT_SIZE - PARAM_SIZE)  // 7934
#define PF_DIST      8   // prefetch distance in grid-stride iterations

typedef unsigned int v4u __attribute__((ext_vector_type(4)));
typedef int          v8i __attribute__((ext_vector_type(8)));
typedef int          v4i __attribute__((ext_vector_type(4)));
typedef float        v2f __attribute__((ext_vector_type(2)));
typedef float        v8f __attribute__((ext_vector_type(8)));

// ---------------------------------------------------------------------------
// Workspace layout (d_ws):
//   [0 .. 131072)            float acc[HIDDEN*BATCH]   (acc[h*16 + b])
//   [131072 .. 147456)       float stage[1024*16]      (inputs^T, stage[c*16+b])
// Output layout (d_out, floats):
//   [0 .. 4096)              outputs [16][256]
//   [4096 .. 135168)         new_x   [16][8192]
//   [135168]                 energy scalar
// ---------------------------------------------------------------------------

__global__ void __launch_bounds__(256)
prep_kernel(const float* __restrict__ inputs,
            float* __restrict__ acc,
            float* __restrict__ stage,
            float* __restrict__ energy_slot) {
    const int stride = (int)(gridDim.x * blockDim.x);
    const int n = BATCH * HIDDEN_SIZE;
    for (int t = (int)(blockIdx.x * blockDim.x + threadIdx.x); t < n; t += stride) {
        acc[t] = 0.0f;
        if (t < INPUT_SIZE * BATCH) {
            const int c = t >> 4;
            const int b = t & 15;
            stage[t] = inputs[b * INPUT_SIZE + c];   // transpose: stage[c*16+b]
        }
    }
    if (blockIdx.x == 0 && threadIdx.x == 0) *energy_slot = 0.0f;
}

__global__ void __launch_bounds__(256)
spmm_kernel(const float* __restrict__ wv,
            const int*   __restrict__ rows,
            const int*   __restrict__ cols,
            const float* __restrict__ y,
            const float* __restrict__ stage_global,
            float* __restrict__ acc,
            int nnz) {
    __shared__ float lds_in[INPUT_SIZE * BATCH];   // 64 KB staged via TDM

    // --- Tensor Data Mover: DMA the 64KB inputs^T tile into LDS (wave 0) ---
    if (threadIdx.x < 32) {
        const uint64_t ga       = (uint64_t)(uintptr_t)stage_global;
        const uint32_t lds_base = (uint32_t)(uintptr_t)lds_in;    // LDS byte offset
        const uint32_t N        = (uint32_t)(INPUT_SIZE * BATCH); // 16384 elems, 4B

        v4u g0;
        g0.x = 1u;                                          // count=1 (valid), user mode
        g0.y = lds_base;                                    // lds_addr
        g0.z = (uint32_t)(ga & 0xFFFFFFFFu);                // global_addr[31:0]
        g0.w = (uint32_t)((ga >> 32) & 0x01FFFFFFu)         // global_addr[56:32]
             | (2u << 30);                                  // type = 2 ("image")

        v8i g1;
        g1[0] = (int)(2u << 16);                            // data_size = 4 bytes
        g1[1] = (int)((N & 0xFFFFu) << 16);                 // tensor_dim0[15:0]
        g1[2] = (int)(((N >> 16) & 0xFFFFu) | (1u << 16));  // tensor_dim0[31:16] | tensor_dim1=1
        g1[3] = (int)((N & 0xFFFFu) << 16);                 // tile_dim0 = 16384
        g1[4] = 1;                                          // tile_dim1 = 1
        g1[5] = (int)N;                                     // tensor_dim0_stride[31:0]
        g1[6] = 0;
        g1[7] = 0;

        v4i z4 = {0, 0, 0, 0};                              // groups 2/3 unused (<=2D)
        v8i z8 = {0, 0, 0, 0, 0, 0, 0, 0};
        __builtin_amdgcn_tensor_load_to_lds(g0, g1, z4, z4, z8, 0);
        __builtin_amdgcn_s_wait_tensorcnt(0);
    }
    __syncthreads();

    // --- 16 lanes per nonzero: lane b handles batch element b ---
    const int lane16  = (int)(threadIdx.x & 15);
    const int group   = (int)((blockIdx.x * blockDim.x + threadIdx.x) >> 4);
    const int ngroups = (int)((gridDim.x * blockDim.x) >> 4);

    for (int e = group; e < nnz; e += ngroups) {
        // speculative prefetch of the streaming triple (global_prefetch_b8);
        // translation failures past the end are silently dropped
        const int ep = e + PF_DIST * ngroups;
        __builtin_prefetch(&wv[ep],   0, 1);
        __builtin_prefetch(&rows[ep], 0, 1);
        __builtin_prefetch(&cols[ep], 0, 1);

        const float w = wv[e];
        const int   r = rows[e];
        const int   c = cols[e];
        float v;
        if (c < INPUT_SIZE) {
            v = lds_in[(c << 4) + lane16];       // ds_load, 16 consecutive banks
        } else {
            v = y[c - INPUT_SIZE];               // broadcast, L2-hot (32 KB table)
        }
        atomicAdd(&acc[(r << 4) + lane16], w * v);
    }
}

__global__ void __launch_bounds__(256)
finalize_kernel(const float* __restrict__ acc,
                const float* __restrict__ x,
                const float* __restrict__ bias,
                const float* __restrict__ tau,
                const int*   __restrict__ act_type,
                float* __restrict__ out_outputs,   // [16][256]
                float* __restrict__ out_newx,      // [16][8192]
                float* __restrict__ energy_slot) {
    const int i = (int)(blockIdx.x * blockDim.x + threadIdx.x); // [0, 131072)
    const int b = i >> 13;                    // batch
    const int h = i & (HIDDEN_SIZE - 1);      // neuron

    const float xv = x[h];
    const float a  = acc[(h << 4) + b];
    const float nx = xv + ((-xv + a + bias[h]) / tau[h]) * DT_STEP;
    out_newx[i] = nx;                         // i == b*8192 + h, coalesced

    const int t = act_type[h];
    float ny;
    if (t == 0)       ny = tanhf(nx);
    else if (t == 1)  ny = 1.0f / (1.0f + expf(-nx));
    else              ny = (nx > 0.0f) ? nx : 0.0f;

    if (h >= LO_IDX && h < LO_IDX + OUTPUT_SIZE)
        out_outputs[b * OUTPUT_SIZE + (h - LO_IDX)] = ny;

    // --- energy: wave32 reduce (WMMA reduction-by-ones) -> LDS -> atomic ---
    const float val = fabsf(ny);
    float p;
#if defined(__has_builtin) && __has_builtin(__builtin_amdgcn_wmma_f32_16x16x4_f32)
    {
        // A (16x4 f32, 2 VGPRs): A[m][0] = val(lane m), A[m][2] = val(lane m+16),
        // A[m][1] = A[m][3] = 0.  B = ones(4x16).  D[m][n] = val[m] + val[m+16].
        v2f am; am.x = val;  am.y = 0.0f;
        v2f bm; bm.x = 1.0f; bm.y = 1.0f;
        v8f cm = {};
        v8f d = __builtin_amdgcn_wmma_f32_16x16x4_f32(
            false, am, false, bm, (short)0, cm, false, false);
        // lanes 0-15 hold rows 0-7, lanes 16-31 hold rows 8-15 (VGPRs 0..7)
        p = d[0] + d[1] + d[2] + d[3] + d[4] + d[5] + d[6] + d[7];
        p += __shfl_xor(p, 16, 32);          // combine the two row halves
    }
#else
    p = val;
    #pragma unroll
    for (int off = 16; off > 0; off >>= 1)
        p += __shfl_down(p, off, 32);
#endif

    __shared__ float red[8];
    const int wid = (int)(threadIdx.x >> 5);
    if ((threadIdx.x & 31) == 0) red[wid] = p;
    __syncthreads();
    if (threadIdx.x == 0) {
        float s = 0.0f;
        #pragma unroll
        for (int k = 0; k < 8; ++k) s += red[k];
        atomicAdd(energy_slot, s * 0.001f);
    }
}

extern "C" void kernel_launch(void* const* d_in, const int* in_sizes, int n_in,
                              void* d_out, int out_size, void* d_ws, size_t ws_size,
                              hipStream_t stream) {
    const float* inputs   = (const float*)d_in[0];
    const float* x        = (const float*)d_in[1];
    const float* y        = (const float*)d_in[2];
    const float* tau      = (const float*)d_in[3];
    const float* bias     = (const float*)d_in[4];
    const float* wv       = (const float*)d_in[5];
    const int*   rows     = (const int*)d_in[6];
    const int*   cols     = (const int*)d_in[7];
    const int*   act_type = (const int*)d_in[8];
    const int    nnz      = in_sizes[5];

    float* acc   = (float*)d_ws;                         // 131072 floats
    float* stage = acc + BATCH * HIDDEN_SIZE;            // 16384 floats

    float* out_outputs = (float*)d_out;                  // 4096 floats
    float* out_newx    = out_outputs + BATCH * OUTPUT_SIZE;  // 131072 floats
    float* energy_slot = out_newx + BATCH * HIDDEN_SIZE;     // 1 float

    prep_kernel<<<512, 256, 0, stream>>>(inputs, acc, stage, energy_slot);
    spmm_kernel<<<1024, 256, 0, stream>>>(wv, rows, cols, y, stage, acc, nnz);
    finalize_kernel<<<512, 256, 0, stream>>>(acc, x, bias, tau, act_type,
                                             out_outputs, out_newx, energy_slot);
}